// AttentionPooler_14328010900074
// MI455X (gfx1250) — compile-verified
//
#include <hip/hip_runtime.h>

// CDNA5 / gfx1250, wave32. f32 WMMA: D(16x16,f32) = A(16x4,f32) x B(4x16,f32) + C
typedef __attribute__((ext_vector_type(2))) float v2f;
typedef __attribute__((ext_vector_type(8))) float v8f;

#define WMMA_F32X4(a, b, c) \
    __builtin_amdgcn_wmma_f32_16x16x4_f32(false, (a), false, (b), (short)0, (c), false, false)

// Problem constants (match reference)
#define BB 32   // batch
#define NN 512  // ragged items
#define SS 32   // static input size
#define WW 64   // width (K of GEMM1, also cols of W1)
#define PP 64   // layer_param (cols of W2)

// ---------------------------------------------------------------------------
// zero the h-sum accumulator in workspace (B*S x W = 1024 x 64 f32 = 256 KB)
// ---------------------------------------------------------------------------
__global__ void zero_f32(float* __restrict__ p, int n) {
    int i = blockIdx.x * blockDim.x + threadIdx.x;
    if (i < n) p[i] = 0.0f;
}

// ---------------------------------------------------------------------------
// GEMM1 fused with bias + ReLU + reduction over N:
//   hacc[b,st*16+m, j] += sum_n relu(X[b,n,s,:] @ W1 + b1)[j]
// grid: B*2*CHUNKS workgroups (CHUNKS=8), 128 threads (4 waves) each.
// Each wave owns 16 consecutive n values; one token tile = 16 s-rows x 64.
// ---------------------------------------------------------------------------
__global__ __launch_bounds__(128) void gemm1_relu_nsum(
    const float* __restrict__ X,   // [B,N,S,W] f32
    const float* __restrict__ W1,  // [W,W] row-major
    const float* __restrict__ b1,  // [W]
    float* __restrict__ hacc)      // [B*2][16][64] f32 accumulator
{
    // LDS-staged W1 lane-fragments: bsh[c][kk][lane] = {W1[k0, col], W1[k0+1, col]}
    // col = 16*c + lane%16 ; k0 = 4*kk + 2*(lane/16)
    __shared__ v2f bsh[4][16][32];

    const int t = threadIdx.x;
    for (int i = t; i < 4 * 16 * 32; i += 128) {
        int c   = i >> 9;          // /512
        int kk  = (i >> 5) & 15;
        int l   = i & 31;
        int col = c * 16 + (l & 15);
        int k0  = kk * 4 + ((l >> 4) << 1);
        v2f v;
        v.x = W1[k0 * WW + col];
        v.y = W1[(k0 + 1) * WW + col];
        bsh[c][kk][l] = v;
    }
    __syncthreads();

    const int wg    = blockIdx.x;
    const int chunk = wg & 7;          // 8 n-chunks of 64
    const int bs    = wg >> 3;         // b*2 + st  (st = 16-row s-tile)
    const int wave  = t >> 5;
    const int lane  = t & 31;
    const int n0    = chunk * 64 + wave * 16;   // this wave: n0 .. n0+15

    const int b  = bs >> 1;
    const int st = bs & 1;

    // A-fragment lane offset inside a 16x64 tile (elements)
    const int rowoff = (lane & 15) * WW + ((lane >> 4) << 1);
    const float* base0 = X + (size_t)b * NN * SS * WW + (size_t)st * 16 * WW;

    v8f hsum[4];
    #pragma unroll
    for (int c = 0; c < 4; ++c) hsum[c] = (v8f){};

    for (int tile = 0; tile < 16; ++tile) {
        const float* base = base0 + (size_t)(n0 + tile) * (SS * WW);
        v2f a[16];
        #pragma unroll
        for (int kk = 0; kk < 16; ++kk)
            a[kk] = *(const v2f*)(base + rowoff + 4 * kk);

        #pragma unroll
        for (int c = 0; c < 4; ++c) {
            v8f acc = (v8f){};
            #pragma unroll
            for (int kk = 0; kk < 16; ++kk)
                acc = WMMA_F32X4(a[kk], bsh[c][kk][lane], acc);

            const float bias = b1[c * 16 + (lane & 15)];
            #pragma unroll
            for (int e = 0; e < 8; ++e) {
                float v = acc[e] + bias;
                hsum[c][e] += (v > 0.0f) ? v : 0.0f;
            }
        }
    }

    // accumulate partial h-sums into global accumulator
    float* dst = hacc + (size_t)bs * (16 * WW);
    #pragma unroll
    for (int c = 0; c < 4; ++c) {
        const int j = c * 16 + (lane & 15);
        #pragma unroll
        for (int e = 0; e < 8; ++e) {
            const int m = e + ((lane >> 4) << 3);
            atomicAdd(dst + m * WW + j, hsum[c][e]);
        }
    }
}

// ---------------------------------------------------------------------------
// GEMM2: out[b, st*16+m, p] = hacc[bs, m, :] @ W2 + N*b2
// grid: 64 workgroups, 1 wave each.
// ---------------------------------------------------------------------------
__global__ __launch_bounds__(32) void gemm2_bias(
    const float* __restrict__ hacc, // [B*2][16][64]
    const float* __restrict__ W2,   // [W,P] row-major
    const float* __restrict__ b2,   // [P]
    float* __restrict__ out)        // [B,S,P]
{
    const int bs   = blockIdx.x;
    const int lane = threadIdx.x & 31;
    const int b    = bs >> 1;
    const int st   = bs & 1;

    const int rowoff = (lane & 15) * WW + ((lane >> 4) << 1);
    const float* base = hacc + (size_t)bs * (16 * WW);

    v2f a[16];
    #pragma unroll
    for (int kk = 0; kk < 16; ++kk)
        a[kk] = *(const v2f*)(base + rowoff + 4 * kk);

    #pragma unroll
    for (int c = 0; c < 4; ++c) {
        const int col = c * 16 + (lane & 15);
        v8f acc = (v8f){};
        #pragma unroll
        for (int kk = 0; kk < 16; ++kk) {
            const int k0 = 4 * kk + ((lane >> 4) << 1);
            v2f bf;
            bf.x = W2[k0 * PP + col];
            bf.y = W2[(k0 + 1) * PP + col];
            acc = WMMA_F32X4(a[kk], bf, acc);
        }
        const float bias = (float)NN * b2[col];
        #pragma unroll
        for (int e = 0; e < 8; ++e) {
            const int m = e + ((lane >> 4) << 3);
            out[((size_t)(b * SS + st * 16 + m)) * PP + col] = acc[e] + bias;
        }
    }
}

// ---------------------------------------------------------------------------
extern "C" void kernel_launch(void* const* d_in, const int* in_sizes, int n_in,
                              void* d_out, int out_size, void* d_ws, size_t ws_size,
                              hipStream_t stream) {
    const float* X  = (const float*)d_in[0];
    const float* W1 = (const float*)d_in[1];
    const float* b1 = (const float*)d_in[2];
    const float* W2 = (const float*)d_in[3];
    const float* b2 = (const float*)d_in[4];
    float* out  = (float*)d_out;
    float* hacc = (float*)d_ws;           // 64 * 16 * 64 f32 = 256 KB

    const int haccN = BB * 2 * 16 * WW;   // 65536
    zero_f32<<<(haccN + 255) / 256, 256, 0, stream>>>(hacc, haccN);
    gemm1_relu_nsum<<<BB * 2 * 8, 128, 0, stream>>>(X, W1, b1, hacc);
    gemm2_bias<<<BB * 2, 32, 0, stream>>>(hacc, W2, b2, out);
}